// Diffusion_27530740367509
// MI455X (gfx1250) — compile-verified
//
#include <hip/hip_runtime.h>

// ---------------------------------------------------------------------------
// Heat diffusion on a graph: out = expm(-t*L) @ x, scaled Taylor series.
//   N = 50000 nodes, C = 64 channels, 850k COO entries, 4 substeps x 20 terms.
// Structure per Taylor term k:
//   tmp = L @ term            (atomic scatter over edge list, tmp pre-zeroed)
//   term = (scale/k) * tmp    \  fused in one WMMA kernel that also
//   out  += term              /  re-zeros tmp for the next scatter
// scale = -(clamp(t,1e-8)/4)
//
// Everything (x, edges, 2 scratch buffers, out) is ~61 MB -> resident in the
// 192 MB L2, so the SpMM is L2-gather/atomic bound. f32 throughout (Taylor
// cancellation forbids low precision). WMMA f32_16x16x4 does the exact scaled
// accumulate via column-selector B matrices (values are c or 0, so each output
// element is a single rounded product: bit-identical to a VALU c*v).
// ---------------------------------------------------------------------------

typedef float v2f __attribute__((ext_vector_type(2)));
typedef float v8f __attribute__((ext_vector_type(8)));

#define CCH 64  // channels

// out = x ; term = x ; tmp = 0
__global__ void diff_init(const float* __restrict__ x, float* __restrict__ out,
                          float* __restrict__ term, float* __restrict__ tmp, int n) {
    int i = blockIdx.x * blockDim.x + threadIdx.x;
    if (i < n) {
        float v = x[i];
        out[i] = v; term[i] = v; tmp[i] = 0.0f;
    }
}

// term = out ; tmp = 0   (start of substeps 2..4)
__global__ void diff_resub(const float* __restrict__ out, float* __restrict__ term,
                           float* __restrict__ tmp, int n) {
    int i = blockIdx.x * blockDim.x + threadIdx.x;
    if (i < n) { term[i] = out[i]; tmp[i] = 0.0f; }
}

// One wave per edge: lane l handles channels 2l, 2l+1.
// src/dst/w are wave-uniform -> forced scalar (s_load) via readfirstlane.
__global__ void spmm_scatter(const int* __restrict__ esrc, const int* __restrict__ edst,
                             const float* __restrict__ ew, const float* __restrict__ term,
                             float* __restrict__ tmp, int nE) {
    int lane = threadIdx.x & 31;
    int wid  = blockIdx.x * (blockDim.x >> 5) + (threadIdx.x >> 5);
    int e = __builtin_amdgcn_readfirstlane(wid);
    if (e >= nE) return;
    int   s = esrc[e];          // uniform address -> SMEM load
    int   d = edst[e];
    float w = ew[e];
    const v2f* vrow = (const v2f*)(term + (size_t)d * CCH);
    v2f v = vrow[lane];         // 256 B coalesced gather per wave (L2 hit)
    float* orow = tmp + (size_t)s * CCH;
    atomicAdd(orow + 2 * lane,     w * v.x);   // global_atomic_add_f32 (no rtn)
    atomicAdd(orow + 2 * lane + 1, w * v.y);
}

// One wave per 16x16 tile (16 nodes x 16 channels).
//   d = sum_j A_j x B_j  (+0)  == coef * tmp_tile   (exact: selector B)
//   term_tile = d ; out_tile += d ; tmp_tile = 0
__global__ void fused_axpy_wmma(float* __restrict__ tmp, float* __restrict__ term,
                                float* __restrict__ out, const float* __restrict__ tptr,
                                int k, int nNodes) {
    const int lane = threadIdx.x & 31;
    const int tile = blockIdx.x * (blockDim.x >> 5) + (threadIdx.x >> 5);
    const int nTiles = ((nNodes + 15) >> 4) * (CCH / 16);
    if (tile >= nTiles) return;                  // wave-uniform branch
    const int nb = (tile / (CCH / 16)) * 16;     // node-block base
    const int cb = (tile % (CCH / 16)) * 16;     // channel-block base

    float tt = tptr[0];
    if (tt < 1e-8f) tt = 1e-8f;                  // clamp(t, 1e-8)
    const float coef = -(tt * 0.25f) / (float)k; // scale/k, scale=-t/4

    const int hi = lane >> 4;    // half-wave select
    const int lr = lane & 15;

    if (nb + 16 <= nNodes) {
        // --- WMMA path (EXEC all-1s: branches above are wave-uniform) ---
        v8f d = {};
#pragma unroll
        for (int j = 0; j < 4; ++j) {
            // A_j (16x4 f32 layout): lane -> row lr, VGPR p -> K = p + 2*hi
            const float* ap = tmp + (size_t)(nb + lr) * CCH + cb + 4 * j + 2 * hi;
            v2f a = *(const v2f*)ap;             // 8B-aligned pair load
            // B_j (4x16 selector): B(K,N) = coef iff N == 4j+K
            v2f b;
            b.x = (lr == 4 * j + 2 * hi + 0) ? coef : 0.0f;
            b.y = (lr == 4 * j + 2 * hi + 1) ? coef : 0.0f;
            d = __builtin_amdgcn_wmma_f32_16x16x4_f32(
                    false, a, false, b, (short)0, d, false, false);
        }
        // C/D layout: VGPR r, lane -> element (M = r + 8*hi, N = lr)
        const size_t base = (size_t)(nb + 8 * hi) * CCH + cb + lr;
        float o[8];
#pragma unroll
        for (int r = 0; r < 8; ++r) o[r] = out[base + (size_t)r * CCH];
        // all tile loads (A_j by sibling lanes too) must land before we
        // zero-fill tmp below
        asm volatile("s_wait_loadcnt 0" ::: "memory");
#pragma unroll
        for (int r = 0; r < 8; ++r) {
            size_t off = base + (size_t)r * CCH;
            out[off]  = o[r] + d[r];
            term[off] = d[r];
            tmp[off]  = 0.0f;
        }
    } else {
        // --- scalar remainder tile (not hit for N % 16 == 0) ---
        for (int idx = lane; idx < 256; idx += 32) {
            int m = nb + (idx >> 4);
            int n = cb + (idx & 15);
            if (m < nNodes) {
                size_t off = (size_t)m * CCH + n;
                float tv = coef * tmp[off];
                term[off] = tv;
                out[off] += tv;
                tmp[off] = 0.0f;
            }
        }
    }
}

extern "C" void kernel_launch(void* const* d_in, const int* in_sizes, int n_in,
                              void* d_out, int out_size, void* d_ws, size_t ws_size,
                              hipStream_t stream) {
    const float* x    = (const float*)d_in[0];
    const int*   esrc = (const int*)d_in[1];
    const int*   edst = (const int*)d_in[2];
    const float* ew   = (const float*)d_in[3];
    const float* t    = (const float*)d_in[4];

    const int NC     = in_sizes[0];       // N * 64
    const int nNodes = NC / CCH;
    const int nE     = in_sizes[1];       // 850000

    float* out  = (float*)d_out;
    float* term = (float*)d_ws;           // N*C floats
    float* tmp  = term + (size_t)NC;      // N*C floats (ws needs 2*N*C*4 bytes)

    const dim3 blk(256);
    const int gridNC = (NC + 255) / 256;
    const int gridE  = (nE + 7) / 8;                         // 8 waves/block, 1 edge/wave
    const int nTiles = ((nNodes + 15) / 16) * (CCH / 16);
    const int gridT  = (nTiles + 7) / 8;                     // 8 tiles/block

    diff_init<<<gridNC, blk, 0, stream>>>(x, out, term, tmp, NC);

    for (int s = 0; s < 4; ++s) {                            // N_SUB substeps
        if (s) diff_resub<<<gridNC, blk, 0, stream>>>(out, term, tmp, NC);
        for (int k = 1; k <= 20; ++k) {                      // Taylor terms
            spmm_scatter<<<gridE, blk, 0, stream>>>(esrc, edst, ew, term, tmp, nE);
            fused_axpy_wmma<<<gridT, blk, 0, stream>>>(tmp, term, out, t, k, nNodes);
        }
    }
}